// MiniMaxText01LinearAttention_89687507076311
// MI455X (gfx1250) — compile-verified
//
#include <hip/hip_runtime.h>
#include <math.h>

#define SEQ   4096
#define HID   2048
#define NH    32
#define HD    128
#define INNER 4096
#define BLK   256
#define NB    (SEQ / BLK)

typedef __attribute__((ext_vector_type(2))) float v2f;
typedef __attribute__((ext_vector_type(8))) float v8f;

// ---- CDNA5 async global->LDS staging (guarded: falls back to float4 copies) ----
#if defined(__has_builtin)
#if __has_builtin(__builtin_amdgcn_global_load_async_to_lds_b128) && \
    __has_builtin(__builtin_amdgcn_s_wait_asynccnt)
#define USE_ASYNC_LDS 1
#endif
#endif

typedef int v4i __attribute__((vector_size(16)));
typedef __attribute__((address_space(1))) v4i* gv4p;  // global 16B vector ptr
typedef __attribute__((address_space(3))) v4i* lv4p;  // LDS 16B vector ptr

// copy 16B from global to LDS (per-lane addresses)
__device__ __forceinline__ void cp_g2l_16B(float* l, const float* g) {
#ifdef USE_ASYNC_LDS
  __builtin_amdgcn_global_load_async_to_lds_b128((gv4p)g, (lv4p)l, 0, 0);
#else
  *(float4*)l = *(const float4*)g;
#endif
}

__device__ __forceinline__ void cp_wait() {
#ifdef USE_ASYNC_LDS
  __builtin_amdgcn_s_wait_asynccnt(0);
#endif
}

__device__ __forceinline__ v8f wmma4(v2f a, v2f b, v8f c) {
  // D = A(16x4,f32) * B(4x16,f32) + C(16x16,f32)
  return __builtin_amdgcn_wmma_f32_16x16x4_f32(false, a, false, b, (short)0, c, false, false);
}

__device__ __forceinline__ v8f vzero8() {
  v8f z = {0.f, 0.f, 0.f, 0.f, 0.f, 0.f, 0.f, 0.f};
  return z;
}

// A-fragment (16x4) from row-major matrix: lane m=lane%16 holds row m,
// VGPR pair = K {2*(lane/16), 2*(lane/16)+1}  -> contiguous 8B load.
__device__ __forceinline__ v2f afrag(const float* base, int ld, int row0, int k0, int lane) {
  int m = lane & 15, g = lane >> 4;
  const float* p = base + (size_t)(row0 + m) * ld + (k0 + 2 * g);
  v2f a;
  a.x = p[0];
  a.y = p[1];
  return a;
}

// B-fragment (4x16) from row-major matrix: lane n=lane%16 holds column n,
// VGPR pair = K rows {2*(lane/16), 2*(lane/16)+1}.
__device__ __forceinline__ v2f bfrag(const float* base, int ld, int k0, int col0, int lane) {
  int n = lane & 15, g = lane >> 4;
  const float* p = base + (size_t)(k0 + 2 * g) * ld + (col0 + n);
  v2f b;
  b.x = p[0];
  b.y = p[ld];
  return b;
}

// ---------------------------------------------------------------------------
// Kernel 1: qkv = silu(X @ Wqkv) scattered to [h][n][d];  gate = sigmoid(X @ Wgate)
// grid = (SEQ/128, 96+32), block = 256 (8 waves, each 32x64 of a 128x128 tile)
// ---------------------------------------------------------------------------
__launch_bounds__(256, 2)
__global__ void gemm_qkv_gate(const float* __restrict__ X, const float* __restrict__ Wqkv,
                              const float* __restrict__ Wgate, float* __restrict__ ws) {
  __shared__ float la[128 * 36];   // A tile 128x32, padded stride 36 (bank-conflict-free A-frags)
  __shared__ float lb[32 * 128];   // B tile 32x128
  const int mt = blockIdx.x, nt = blockIdx.y;
  const int t = threadIdx.x, w = t >> 5, lane = t & 31;
  const int wm = w >> 1, wn = w & 1;  // 4x2 wave grid
  const int m0 = mt * 128;

  const float* B;
  int ldb, bn0;
  bool isGate;
  if (nt < 96) { B = Wqkv;  ldb = 3 * INNER; bn0 = nt * 128;        isGate = false; }
  else         { B = Wgate; ldb = INNER;     bn0 = (nt - 96) * 128; isGate = true;  }

  v8f acc[2][4];
#pragma unroll
  for (int i = 0; i < 2; ++i)
#pragma unroll
    for (int j = 0; j < 4; ++j) acc[i][j] = vzero8();

  for (int ks = 0; ks < HID / 32; ++ks) {
#pragma unroll
    for (int i = 0; i < 4; ++i) {
      int idx = t + i * 256;
      int r = idx >> 3, c = (idx & 7) * 4;
      cp_g2l_16B(la + r * 36 + c, X + (size_t)(m0 + r) * HID + ks * 32 + c);
      int rb = idx >> 5, cb = (idx & 31) * 4;
      cp_g2l_16B(lb + rb * 128 + cb, B + (size_t)(ks * 32 + rb) * ldb + bn0 + cb);
    }
    if (ks + 1 < HID / 32) {
      __builtin_prefetch(X + (size_t)(m0 + (t >> 3)) * HID + (ks + 1) * 32, 0, 1);
      __builtin_prefetch(B + (size_t)((ks + 1) * 32 + (t >> 5)) * ldb + bn0, 0, 1);
    }
    cp_wait();
    __syncthreads();
#pragma unroll
    for (int kk = 0; kk < 8; ++kk) {
      v2f a0 = afrag(la, 36, wm * 32, kk * 4, lane);
      v2f a1 = afrag(la, 36, wm * 32 + 16, kk * 4, lane);
      v2f bf[4];
#pragma unroll
      for (int ni = 0; ni < 4; ++ni) bf[ni] = bfrag(lb, 128, kk * 4, wn * 64 + ni * 16, lane);
#pragma unroll
      for (int ni = 0; ni < 4; ++ni) {
        acc[0][ni] = wmma4(a0, bf[ni], acc[0][ni]);
        acc[1][ni] = wmma4(a1, bf[ni], acc[1][ni]);
      }
    }
    __syncthreads();
  }

  const int n = lane & 15, g = lane >> 4;
  float* qws = ws;
  float* kws = ws + (size_t)NH * SEQ * HD;
  float* vws = ws + 2 * (size_t)NH * SEQ * HD;
  float* gws = ws + 3 * (size_t)NH * SEQ * HD;

#pragma unroll
  for (int mi = 0; mi < 2; ++mi)
#pragma unroll
    for (int ni = 0; ni < 4; ++ni)
#pragma unroll
      for (int r = 0; r < 8; ++r) {
        int row = m0 + wm * 32 + mi * 16 + r + 8 * g;
        int colLocal = wn * 64 + ni * 16 + n;
        float x = acc[mi][ni][r];
        if (isGate) {
          gws[(size_t)row * INNER + bn0 + colLocal] = 1.f / (1.f + __expf(-x));
        } else {
          float sl = x / (1.f + __expf(-x));  // silu
          int c = bn0 + colLocal;             // 0..12287
          int head = c / 384;
          int part = (c % 384) / 128;         // 0=q 1=k 2=v
          int dd = c & 127;
          float* dst = (part == 0) ? qws : (part == 1) ? kws : vws;
          dst[(size_t)head * SEQ * HD + (size_t)row * HD + dd] = sl;
        }
      }
}

// ---------------------------------------------------------------------------
// Kernel 2: lightning attention. One workgroup (512 thr = 16 waves) per head.
// Wave w owns rows [16w,16w+16) of each 256-row block; waves 0..7 own KV strips.
// ---------------------------------------------------------------------------
__launch_bounds__(512, 1)
__global__ void lightning_attn(const float* __restrict__ ws_qkv, const float* __restrict__ slope,
                               const float* __restrict__ kv0, float* __restrict__ attn) {
  __shared__ float lk[BLK * HD];       // 128 KB : K block (256x128)
  __shared__ float lv[128 * HD];       // 64 KB  : V half-block (128x128)
  __shared__ float lkv[HD * HD];       // 64 KB  : KV state (128x128)
  __shared__ float lsc[16 * 16 * 20];  // 20 KB  : per-wave S tile scratch (stride 20)

  const int h = blockIdx.x;
  const int t = threadIdx.x, w = t >> 5, lane = t & 31;
  const int n = lane & 15, g = lane >> 4;
  const float s = slope[h];
  const float* qh = ws_qkv + (size_t)h * SEQ * HD;
  const float* kh = ws_qkv + (size_t)NH * SEQ * HD + (size_t)h * SEQ * HD;
  const float* vh = ws_qkv + 2 * (size_t)NH * SEQ * HD + (size_t)h * SEQ * HD;
  const float bdecay = __expf(-s * (float)BLK);
  const int i0 = w * 16;

  for (int i = t; i < HD * HD; i += 512) lkv[i] = kv0[(size_t)h * HD * HD + i];
  __syncthreads();

  for (int b = 0; b < NB; ++b) {
    __syncthreads();  // prior lk reads / lkv writes done
#pragma unroll
    for (int i = 0; i < 16; ++i) {  // stage K block: 256x128
      int idx = t + i * 512;
      int r = idx >> 5, c = (idx & 31) * 4;
      cp_g2l_16B(lk + r * HD + c, kh + (size_t)(b * BLK + r) * HD + c);
    }
    cp_wait();
    __syncthreads();

    // q fragments for this wave's 16-row strip (reused for scores and o_inter)
    const float* qB = qh + (size_t)(b * BLK) * HD;
    v2f qA[32];
#pragma unroll
    for (int kk = 0; kk < 32; ++kk) qA[kk] = afrag(qB, HD, i0, kk * 4, lane);

    // o_inter = q_decay * (q @ KV_old)
    v8f O[8];
#pragma unroll
    for (int nt2 = 0; nt2 < 8; ++nt2) {
      v8f a = vzero8();
#pragma unroll
      for (int kk = 0; kk < 32; ++kk)
        a = wmma4(qA[kk], bfrag(lkv, HD, kk * 4, nt2 * 16, lane), a);
      O[nt2] = a;
    }
#pragma unroll
    for (int r = 0; r < 8; ++r) {
      float qd = __expf(-s * (float)(i0 + r + 8 * g + 1));
#pragma unroll
      for (int nt2 = 0; nt2 < 8; ++nt2) O[nt2][r] *= qd;
    }

    // KV accumulators: strip rows [16w,16w+16), init = block_decay * KV_old
    v8f KV[8];
    if (w < 8) {
#pragma unroll
      for (int d2 = 0; d2 < 8; ++d2)
#pragma unroll
        for (int r = 0; r < 8; ++r)
          KV[d2][r] = bdecay * lkv[(size_t)(w * 16 + r + 8 * g) * HD + d2 * 16 + n];
    }

    for (int half = 0; half < 2; ++half) {
      __syncthreads();
#pragma unroll
      for (int i = 0; i < 8; ++i) {  // stage V half: 128x128
        int idx = t + i * 512;
        int r = idx >> 5, c = (idx & 31) * 4;
        cp_g2l_16B(lv + r * HD + c, vh + (size_t)(b * BLK + half * 128 + r) * HD + c);
      }
      cp_wait();
      __syncthreads();

      // scores (causal lower-triangular j-tiles only) + o_intra
      for (int jt = half * 8; jt < half * 8 + 8; ++jt) {
        if (jt <= w) {
          v8f S = vzero8();
#pragma unroll
          for (int kk = 0; kk < 32; ++kk)
            S = wmma4(qA[kk], bfrag(lk, HD, kk * 4, jt * 16, lane), S);
          // decay mask: S[i][j] *= (i>=j) ? exp(-s*(i-j)) : 0
          int d0 = i0 + 8 * g - jt * 16 - n;
#pragma unroll
          for (int r = 0; r < 8; ++r) {
            int diff = d0 + r;
            S[r] = (diff >= 0) ? S[r] * __expf(-s * (float)diff) : 0.f;
          }
          // D-layout -> A-layout via per-wave LDS scratch
          float* sw = lsc + w * (16 * 20);
#pragma unroll
          for (int r = 0; r < 8; ++r) sw[(r + 8 * g) * 20 + n] = S[r];
          asm volatile("s_wait_dscnt 0x0" ::: "memory");
          v2f sA[4];
#pragma unroll
          for (int k2 = 0; k2 < 4; ++k2) sA[k2] = afrag(sw, 20, 0, k2 * 4, lane);
          int vrow = jt * 16 - half * 128;  // V row local to staged half
#pragma unroll
          for (int nt2 = 0; nt2 < 8; ++nt2)
#pragma unroll
            for (int k2 = 0; k2 < 4; ++k2)
              O[nt2] = wmma4(sA[k2], bfrag(lv, HD, vrow + k2 * 4, nt2 * 16, lane), O[nt2]);
        }
      }

      // KV += (k * k_decay)^T @ V  over this half's 128 rows
      if (w < 8) {
        for (int kk = 0; kk < 32; ++kk) {
          int j0 = half * 128 + kk * 4 + 2 * g;  // block-local row for a.x (a.y = j0+1)
          v2f aK;
          aK.x = lk[(size_t)j0 * HD + w * 16 + n] * __expf(-s * (float)(BLK - 1 - j0));
          aK.y = lk[(size_t)(j0 + 1) * HD + w * 16 + n] * __expf(-s * (float)(BLK - 2 - j0));
#pragma unroll
          for (int d2 = 0; d2 < 8; ++d2)
            KV[d2] = wmma4(aK, bfrag(lv, HD, kk * 4, d2 * 16, lane), KV[d2]);
        }
      }
    }

    // write O: attn[n_global][h*128 + d]
#pragma unroll
    for (int nt2 = 0; nt2 < 8; ++nt2)
#pragma unroll
      for (int r = 0; r < 8; ++r)
        attn[(size_t)(b * BLK + i0 + r + 8 * g) * INNER + h * HD + nt2 * 16 + n] = O[nt2][r];

    __syncthreads();  // all KV_old reads complete before overwrite
    if (w < 8) {
#pragma unroll
      for (int d2 = 0; d2 < 8; ++d2)
#pragma unroll
        for (int r = 0; r < 8; ++r)
          lkv[(size_t)(w * 16 + r + 8 * g) * HD + d2 * 16 + n] = KV[d2][r];
    }
  }
}

// ---------------------------------------------------------------------------
// Kernel 3: RMSNorm * norm_weight * gate (in place over gate buffer)
// ---------------------------------------------------------------------------
__launch_bounds__(256)
__global__ void norm_gate(const float* __restrict__ attn, const float* __restrict__ nw,
                          float* __restrict__ gate_io) {
  __shared__ float red[256];
  const int row = blockIdx.x, t = threadIdx.x;
  const float* a = attn + (size_t)row * INNER;
  float ss = 0.f;
  for (int c = t; c < INNER; c += 256) {
    float x = a[c];
    ss += x * x;
  }
  red[t] = ss;
  __syncthreads();
  for (int off = 128; off > 0; off >>= 1) {
    if (t < off) red[t] += red[t + off];
    __syncthreads();
  }
  float rs = rsqrtf(red[0] * (1.f / (float)INNER) + 1e-5f);
  float* gg = gate_io + (size_t)row * INNER;
  for (int c = t; c < INNER; c += 256) gg[c] = gg[c] * (a[c] * rs * nw[c]);
}

// ---------------------------------------------------------------------------
// Kernel 4: out = G @ Wout   (4096x4096 @ 4096x2048)
// ---------------------------------------------------------------------------
__launch_bounds__(256, 2)
__global__ void gemm_out(const float* __restrict__ G, const float* __restrict__ Wout,
                         float* __restrict__ out) {
  __shared__ float la[128 * 36];
  __shared__ float lb[32 * 128];
  const int mt = blockIdx.x, nt = blockIdx.y;
  const int t = threadIdx.x, w = t >> 5, lane = t & 31;
  const int wm = w >> 1, wn = w & 1;
  const int m0 = mt * 128, n0 = nt * 128;

  v8f acc[2][4];
#pragma unroll
  for (int i = 0; i < 2; ++i)
#pragma unroll
    for (int j = 0; j < 4; ++j) acc[i][j] = vzero8();

  for (int ks = 0; ks < INNER / 32; ++ks) {
#pragma unroll
    for (int i = 0; i < 4; ++i) {
      int idx = t + i * 256;
      int r = idx >> 3, c = (idx & 7) * 4;
      cp_g2l_16B(la + r * 36 + c, G + (size_t)(m0 + r) * INNER + ks * 32 + c);
      int rb = idx >> 5, cb = (idx & 31) * 4;
      cp_g2l_16B(lb + rb * 128 + cb, Wout + (size_t)(ks * 32 + rb) * HID + n0 + cb);
    }
    if (ks + 1 < INNER / 32) {
      __builtin_prefetch(G + (size_t)(m0 + (t >> 3)) * INNER + (ks + 1) * 32, 0, 1);
      __builtin_prefetch(Wout + (size_t)((ks + 1) * 32 + (t >> 5)) * HID + n0, 0, 1);
    }
    cp_wait();
    __syncthreads();
#pragma unroll
    for (int kk = 0; kk < 8; ++kk) {
      v2f a0 = afrag(la, 36, wm * 32, kk * 4, lane);
      v2f a1 = afrag(la, 36, wm * 32 + 16, kk * 4, lane);
      v2f bf[4];
#pragma unroll
      for (int ni = 0; ni < 4; ++ni) bf[ni] = bfrag(lb, 128, kk * 4, wn * 64 + ni * 16, lane);
#pragma unroll
      for (int ni = 0; ni < 4; ++ni) {
        acc[0][ni] = wmma4(a0, bf[ni], acc[0][ni]);
        acc[1][ni] = wmma4(a1, bf[ni], acc[1][ni]);
      }
    }
    __syncthreads();
  }

  const int n = lane & 15, g = lane >> 4;
#pragma unroll
  for (int mi = 0; mi < 2; ++mi)
#pragma unroll
    for (int ni = 0; ni < 4; ++ni)
#pragma unroll
      for (int r = 0; r < 8; ++r) {
        int row = m0 + wm * 32 + mi * 16 + r + 8 * g;
        int col = n0 + wn * 64 + ni * 16 + n;
        out[(size_t)row * HID + col] = acc[mi][ni][r];
      }
}

// ---------------------------------------------------------------------------
extern "C" void kernel_launch(void* const* d_in, const int* in_sizes, int n_in,
                              void* d_out, int out_size, void* d_ws, size_t ws_size,
                              hipStream_t stream) {
  const float* X     = (const float*)d_in[0];  // [4096,2048]
  const float* Wqkv  = (const float*)d_in[1];  // [2048,12288]
  const float* Wgate = (const float*)d_in[2];  // [2048,4096]
  const float* Wout  = (const float*)d_in[3];  // [4096,2048]
  const float* normw = (const float*)d_in[4];  // [4096]
  const float* slope = (const float*)d_in[5];  // [32]
  const float* kv0   = (const float*)d_in[6];  // [32,128,128]
  float* ws = (float*)d_ws;

  const size_t qkv_elems  = (size_t)3 * NH * SEQ * HD;     // q,k,v [h][n][d]
  const size_t gate_elems = (size_t)SEQ * INNER;
  const size_t attn_elems = (size_t)SEQ * INNER;
  if (ws_size < (qkv_elems + gate_elems + attn_elems) * sizeof(float)) return;

  float* gate_ws = ws + qkv_elems;
  float* attn_ws = gate_ws + gate_elems;

  gemm_qkv_gate<<<dim3(SEQ / 128, 96 + INNER / 128), 256, 0, stream>>>(X, Wqkv, Wgate, ws);
  lightning_attn<<<NH, 512, 0, stream>>>(ws, slope, kv0, attn_ws);
  norm_gate<<<SEQ, 256, 0, stream>>>(attn_ws, normw, gate_ws);
  gemm_out<<<dim3(SEQ / 128, HID / 128), 256, 0, stream>>>(gate_ws, Wout, (float*)d_out);
}